// EigenPooling_34213709480483
// MI455X (gfx1250) — compile-verified
//
#include <hip/hip_runtime.h>
#include <hip/hip_bf16.h>
#include <math.h>

// Batched top-16 SVD factor: out = U[:, :16] * diag(S[:16]) = A @ V16
// via Gram-matrix subspace iteration + Rayleigh-Ritz, all matmuls on
// V_WMMA_F32_16X16X4_F32.
//
// B=8, T=256, D=128, RANK=16 -> 2048 independent 128x128 problems,
// one workgroup (8 wave32 waves) per matrix.

#define DD   128
#define RR   16
#define NMAT 2048
#define NITER 8

typedef __attribute__((ext_vector_type(2))) float v2f;
typedef __attribute__((ext_vector_type(8))) float v8f;

static __device__ inline v8f wmma4(v2f a, v2f b, v8f c) {
  // 8 args: (neg_a, A, neg_b, B, c_mod, C, reuse_a, reuse_b)
  return __builtin_amdgcn_wmma_f32_16x16x4_f32(
      false, a, false, b, (short)0, c, false, false);
}

__global__ __launch_bounds__(256, 1)
void EigenPooling_svd_topk(const float* __restrict__ X,
                           float* __restrict__ OUT) {
  __shared__ float As[DD * DD];   // A, row-major            (64 KB)
  __shared__ float Gs[DD * DD];   // G = A^T A               (64 KB)
  __shared__ float Ys[DD * RR];   // current subspace basis  ( 8 KB)
  __shared__ float Zs[DD * RR];   // temp / V                ( 8 KB)
  __shared__ float Hs[RR * RR];   // Rayleigh quotient       ( 1 KB)
  __shared__ float Ws[RR * RR];   // Jacobi eigenvectors     ( 1 KB)
  __shared__ float red[DD];       // reduction scratch

  const int t    = threadIdx.x;
  const int wave = t >> 5;        // 0..7
  const int lane = t & 31;
  const int m    = lane & 15;     // M/N index within tile
  const int h    = lane >> 4;     // lane half

  // ---- Stage A into LDS (coalesced float4) --------------------------------
  {
    const float4* X4  = reinterpret_cast<const float4*>(
        X + (size_t)blockIdx.x * DD * DD);
    float4* As4 = reinterpret_cast<float4*>(As);
    for (int i = t; i < DD * DD / 4; i += 256) As4[i] = X4[i];
  }
  // Deterministic subspace init: block-identity + small dither (rank 16).
  for (int i = t; i < DD * RR; i += 256) {
    int r = i / RR, c = i % RR;
    Ys[i] = (((r & 15) == c) ? 1.0f : 0.0f)
          + 1.0e-3f * (float)((r * 13 + c * 7) & 15);
  }
  __syncthreads();

  // ---- G = A^T A : 8x8 tiles, wave w owns row-tile i=w --------------------
  {
    const int i = wave;
    for (int j = 0; j < 8; ++j) {
      v8f acc = {};
      for (int k = 0; k < 32; ++k) {
        const int kr = k * 4 + 2 * h;
        v2f a, b;
        a.x = As[kr * DD + i * 16 + m];        // A^T[i16+m, kr]   = A[kr, i16+m]
        a.y = As[(kr + 1) * DD + i * 16 + m];
        b.x = As[kr * DD + j * 16 + m];        // A[kr, j16+n], n == m
        b.y = As[(kr + 1) * DD + j * 16 + m];
        acc = wmma4(a, b, acc);
      }
      for (int r = 0; r < 8; ++r)
        Gs[(i * 16 + r + 8 * h) * DD + j * 16 + m] = acc[r];
    }
  }
  __syncthreads();

  // ---- Subspace iteration: Z = G*Y ; MGS(Z) -> Y --------------------------
  for (int it = 0; it <= NITER; ++it) {
    // Z = G * Y   (wave i does row-tile i, K = 128 in steps of 4)
    {
      const int i = wave;
      v8f acc = {};
      for (int k = 0; k < 32; ++k) {
        const int kr = k * 4 + 2 * h;
        v2f a, b;
        a.x = Gs[(i * 16 + m) * DD + kr];
        a.y = Gs[(i * 16 + m) * DD + kr + 1];
        b.x = Ys[kr * RR + m];
        b.y = Ys[(kr + 1) * RR + m];
        acc = wmma4(a, b, acc);
      }
      for (int r = 0; r < 8; ++r)
        Zs[(i * 16 + r + 8 * h) * RR + m] = acc[r];
    }
    __syncthreads();
    if (it == NITER) break;   // final Zs = G*Y kept for Rayleigh-Ritz

    // Modified Gram-Schmidt: orthonormalize columns of Z into Y.
    for (int c = 0; c < RR; ++c) {
      float zc = (t < DD) ? Zs[t * RR + c] : 0.0f;
      for (int p = 0; p < c; ++p) {
        if (t < DD) red[t] = Ys[t * RR + p] * zc;
        __syncthreads();
        for (int s = 64; s > 0; s >>= 1) {
          if (t < s) red[t] += red[t + s];
          __syncthreads();
        }
        const float coef = red[0];
        __syncthreads();
        if (t < DD) zc -= coef * Ys[t * RR + p];
      }
      if (t < DD) red[t] = zc * zc;
      __syncthreads();
      for (int s = 64; s > 0; s >>= 1) {
        if (t < s) red[t] += red[t + s];
        __syncthreads();
      }
      const float inv = rsqrtf(red[0] + 1.0e-30f);
      __syncthreads();
      if (t < DD) Ys[t * RR + c] = zc * inv;
      __syncthreads();
    }
  }

  // ---- Rayleigh-Ritz: H = Y^T * (G*Y)  (single 16x16 tile, wave 0) --------
  if (wave == 0) {
    v8f acc = {};
    for (int k = 0; k < 32; ++k) {
      const int kr = k * 4 + 2 * h;
      v2f a, b;
      a.x = Ys[kr * RR + m];        // Y^T[m, kr] = Y[kr, m]
      a.y = Ys[(kr + 1) * RR + m];
      b.x = Zs[kr * RR + m];
      b.y = Zs[(kr + 1) * RR + m];
      acc = wmma4(a, b, acc);
    }
    for (int r = 0; r < 8; ++r) Hs[(r + 8 * h) * RR + m] = acc[r];
  }
  // init W = I in parallel
  for (int i = t; i < RR * RR; i += 256)
    Ws[i] = ((i / RR) == (i % RR)) ? 1.0f : 0.0f;
  __syncthreads();

  // ---- 16x16 cyclic Jacobi on H, accumulate W; sort desc ------------------
  if (t == 0) {
    for (int sweep = 0; sweep < 6; ++sweep) {
      for (int p = 0; p < RR - 1; ++p) {
        for (int q = p + 1; q < RR; ++q) {
          const float apq = Hs[p * RR + q];
          if (fabsf(apq) < 1.0e-12f) continue;
          const float app = Hs[p * RR + p];
          const float aqq = Hs[q * RR + q];
          const float tau = (aqq - app) / (2.0f * apq);
          const float tt  = (tau >= 0.0f ? 1.0f : -1.0f) /
                            (fabsf(tau) + sqrtf(1.0f + tau * tau));
          const float cc  = rsqrtf(1.0f + tt * tt);
          const float ss  = tt * cc;
          for (int k = 0; k < RR; ++k) {   // rows p,q of H
            const float hp = Hs[p * RR + k], hq = Hs[q * RR + k];
            Hs[p * RR + k] = cc * hp - ss * hq;
            Hs[q * RR + k] = ss * hp + cc * hq;
          }
          for (int k = 0; k < RR; ++k) {   // cols p,q of H and W
            const float hp = Hs[k * RR + p], hq = Hs[k * RR + q];
            Hs[k * RR + p] = cc * hp - ss * hq;
            Hs[k * RR + q] = ss * hp + cc * hq;
            const float wp = Ws[k * RR + p], wq = Ws[k * RR + q];
            Ws[k * RR + p] = cc * wp - ss * wq;
            Ws[k * RR + q] = ss * wp + cc * wq;
          }
        }
      }
    }
    // selection sort eigenpairs (descending eigenvalue), permute W columns
    float ev[RR];
    for (int i2 = 0; i2 < RR; ++i2) ev[i2] = Hs[i2 * RR + i2];
    for (int a = 0; a < RR; ++a) {
      int best = a;
      for (int b2 = a + 1; b2 < RR; ++b2)
        if (ev[b2] > ev[best]) best = b2;
      if (best != a) {
        const float tv = ev[a]; ev[a] = ev[best]; ev[best] = tv;
        for (int k = 0; k < RR; ++k) {
          const float tw = Ws[k * RR + a];
          Ws[k * RR + a]    = Ws[k * RR + best];
          Ws[k * RR + best] = tw;
        }
      }
    }
  }
  __syncthreads();

  // ---- V = Y * W  (128x16 = 8 row-tiles, K = 16) -> Zs --------------------
  {
    const int i = wave;
    v8f acc = {};
    for (int k = 0; k < 4; ++k) {
      const int kr = k * 4 + 2 * h;
      v2f a, b;
      a.x = Ys[(i * 16 + m) * RR + kr];
      a.y = Ys[(i * 16 + m) * RR + kr + 1];
      b.x = Ws[kr * RR + m];
      b.y = Ws[(kr + 1) * RR + m];
      acc = wmma4(a, b, acc);
    }
    __syncthreads();             // done reading old Zs everywhere
    for (int r = 0; r < 8; ++r)
      Zs[(i * 16 + r + 8 * h) * RR + m] = acc[r];
  }
  __syncthreads();

  // ---- L = A * V  ->  OUT  (u_i * s_i = A v_i) ----------------------------
  {
    const int i = wave;
    v8f acc = {};
    for (int k = 0; k < 32; ++k) {
      const int kr = k * 4 + 2 * h;
      v2f a, b;
      a.x = As[(i * 16 + m) * DD + kr];
      a.y = As[(i * 16 + m) * DD + kr + 1];
      b.x = Zs[kr * RR + m];
      b.y = Zs[(kr + 1) * RR + m];
      acc = wmma4(a, b, acc);
    }
    float* outp = OUT + (size_t)blockIdx.x * DD * RR;
    for (int r = 0; r < 8; ++r)
      outp[(i * 16 + r + 8 * h) * RR + m] = acc[r];
  }
}

extern "C" void kernel_launch(void* const* d_in, const int* in_sizes, int n_in,
                              void* d_out, int out_size, void* d_ws, size_t ws_size,
                              hipStream_t stream) {
  (void)in_sizes; (void)n_in; (void)out_size; (void)d_ws; (void)ws_size;
  const float* x = (const float*)d_in[0];   // [8,256,128,128] f32
  float* out = (float*)d_out;               // [8,256,128,16]  f32
  EigenPooling_svd_topk<<<dim3(NMAT), dim3(256), 0, stream>>>(x, out);
}